// PointTransformer_29935922053527
// MI455X (gfx1250) — compile-verified
//
#include <hip/hip_runtime.h>

typedef float v2f __attribute__((ext_vector_type(2)));
typedef float v8f __attribute__((ext_vector_type(8)));
typedef unsigned int u32x4 __attribute__((ext_vector_type(4)));
typedef int i32x4 __attribute__((ext_vector_type(4)));
typedef int i32x8 __attribute__((ext_vector_type(8)));

// ============================================================
// WMMA GEMM:  Y[b,o,j] = sum_c W[o,c] * X(b,c,col(j)) (+bias)(+resid)(relu)
//  - one block = one (batch, 16-row O tile) x 8 column tiles (8 waves)
//  - weight tile staged into LDS once per block by the Tensor Data Mover
//    (TENSOR_LOAD_TO_LDS); TDM OOB rule zero-fills ragged O tiles.
//  - X1 columns optionally gathered through gidx (fused neighbor gather)
//  - X2 (optional) = second channel-concat input, W columns [C1, C1+C2)
//  - arbitrary element strides on X (FC batch-as-columns, broadcast cols)
//  - f32 WMMA (V_WMMA_F32_16X16X4_F32) keeps reference f32 precision.
// ============================================================
__global__ __launch_bounds__(256)
void gemm_wmma_f32(const float* __restrict__ W, const float* __restrict__ bias, int Ctot,
                   const float* __restrict__ X1, int C1, long sb1, long sc1, long sn1,
                   const float* __restrict__ X2, int C2, long sb2, long sc2, long sn2,
                   const int* __restrict__ gidx, long gsb,
                   float* __restrict__ Y, long ysb,
                   const float* __restrict__ resid, long rsb,
                   int O, int Nc, int Bn, int relu, int useLds)
{
  extern __shared__ float ldsW[];                 // [16][Ctot] when useLds
  const int wib = (int)(threadIdx.x >> 5);        // wave in block (0..7)
  const int tiles_n = (Nc + 15) >> 4;
  const int tiles_o = (O + 15) >> 4;
  const int ngroups = (tiles_n + 7) >> 3;
  int bi = (int)blockIdx.x;
  const int b  = bi / (tiles_o * ngroups);
  const int r0 = bi % (tiles_o * ngroups);
  const int ot = r0 / ngroups;
  const int ng = r0 % ngroups;
  const int nt = ng * 8 + wib;

  if (useLds) {
    if (wib == 0) {
      // ---- Tensor DMA: stage W rows [ot*16, ot*16+16) x Ctot into LDS ----
      const unsigned lds_off = (unsigned)(size_t)(void*)ldsW;
      const unsigned long long ga =
          (unsigned long long)(const void*)(W + (long)ot * 16 * Ctot);
      const int rows = O - ot * 16;               // tensor_dim1: OOB rows -> zeros
      u32x4 g0;
      g0.x = 1u;                                   // count=1, user mode
      g0.y = lds_off;                              // lds_addr (bytes)
      g0.z = (unsigned)ga;                         // global_addr[31:0]
      g0.w = (unsigned)((ga >> 32) & 0x1FFFFFFu) | (2u << 30);  // addr[56:32] | type=2
      i32x8 g1;
      g1[0] = (2 << 16);                           // data_size=4B; no multicast/pad
      g1[1] = (int)((Ctot & 0xFFFF) << 16);        // tensor_dim0[15:0]
      g1[2] = (int)(((Ctot >> 16) & 0xFFFF) | ((rows & 0xFFFF) << 16));
      g1[3] = (int)(((rows >> 16) & 0xFFFF) | ((Ctot & 0xFFFF) << 16)); // tile_dim0
      g1[4] = 16;                                  // tile_dim1=16, tile_dim2=0
      g1[5] = Ctot;                                // tensor_dim0_stride[31:0]
      g1[6] = 0;
      g1[7] = 0;
      i32x4 gz = {0, 0, 0, 0};
#if defined(__clang_major__) && (__clang_major__ >= 23)
      i32x8 gz8 = {0, 0, 0, 0, 0, 0, 0, 0};
      __builtin_amdgcn_tensor_load_to_lds(g0, g1, gz, gz, gz8, 0);
#else
      __builtin_amdgcn_tensor_load_to_lds(g0, g1, gz, gz, 0);
#endif
      __builtin_amdgcn_s_wait_tensorcnt((short)0);
    }
    __syncthreads();
  }

  if (nt < tiles_n) {                             // wave-uniform: EXEC stays all-ones
    const int lane = (int)(threadIdx.x & 31);
    const int l = lane & 15, hi = lane >> 4;
    const int orow = ot * 16 + l;
    const int oc = (orow < O) ? orow : (O - 1);   // clamp only used on global-A path
    const int col = nt * 16 + l;
    const int cc = (col < Nc) ? col : (Nc - 1);
    const long scol = gidx ? (long)gidx[(long)b * gsb + cc] : (long)cc;

    const float* Ar = useLds ? (ldsW + (long)l * Ctot) : (W + (long)oc * Ctot);
    v8f acc = {};

    {
      const float* Xb = X1 + (long)b * sb1 + scol * sn1;
      __builtin_prefetch(Xb, 0, 3);
      int c = 0;
      for (; c + 4 <= C1; c += 4) {
        const int kk = c + 2 * hi;
        v2f a, bv;
        a.x  = Ar[kk];
        a.y  = Ar[kk + 1];
        bv.x = Xb[(long)kk * sc1];
        bv.y = Xb[(long)(kk + 1) * sc1];
        acc = __builtin_amdgcn_wmma_f32_16x16x4_f32(false, a, false, bv, (short)0, acc,
                                                    false, false);
      }
      if (c < C1) {                                // K tail (C=3, C=2 cases)
        const int kk = c + 2 * hi;
        v2f a, bv;
        a.x  = (kk     < C1) ? Ar[kk]                   : 0.f;
        a.y  = (kk + 1 < C1) ? Ar[kk + 1]               : 0.f;
        bv.x = (kk     < C1) ? Xb[(long)kk * sc1]       : 0.f;
        bv.y = (kk + 1 < C1) ? Xb[(long)(kk + 1) * sc1] : 0.f;
        acc = __builtin_amdgcn_wmma_f32_16x16x4_f32(false, a, false, bv, (short)0, acc,
                                                    false, false);
      }
    }
    if (X2) {
      const float* Ar2 = Ar + C1;
      const float* Xb = X2 + (long)b * sb2 + (long)cc * sn2;
      int c = 0;
      for (; c + 4 <= C2; c += 4) {
        const int kk = c + 2 * hi;
        v2f a, bv;
        a.x  = Ar2[kk];
        a.y  = Ar2[kk + 1];
        bv.x = Xb[(long)kk * sc2];
        bv.y = Xb[(long)(kk + 1) * sc2];
        acc = __builtin_amdgcn_wmma_f32_16x16x4_f32(false, a, false, bv, (short)0, acc,
                                                    false, false);
      }
      if (c < C2) {
        const int kk = c + 2 * hi;
        v2f a, bv;
        a.x  = (kk     < C2) ? Ar2[kk]                  : 0.f;
        a.y  = (kk + 1 < C2) ? Ar2[kk + 1]              : 0.f;
        bv.x = (kk     < C2) ? Xb[(long)kk * sc2]       : 0.f;
        bv.y = (kk + 1 < C2) ? Xb[(long)(kk + 1) * sc2] : 0.f;
        acc = __builtin_amdgcn_wmma_f32_16x16x4_f32(false, a, false, bv, (short)0, acc,
                                                    false, false);
      }
    }

#pragma unroll
    for (int rr = 0; rr < 8; rr++) {
      const int om = ot * 16 + rr + hi * 8;        // D layout: VGPR rr = M rr / rr+8
      if (om < O && col < Nc) {
        float v = acc[rr];
        if (bias)  v += bias[om];
        if (resid) v += resid[(long)b * rsb + (long)om * Nc + col];
        if (relu)  v = fmaxf(v, 0.f);
        Y[(long)b * ysb + (long)om * Nc + col] = v;
      }
    }
  }
}

// ============================================================
// Brute-force kNN, LDS-tiled source points, stable insertion sort
// (matches top_k tie semantics: equal distance -> lower index first).
// ============================================================
template <int KK>
__global__ __launch_bounds__(256)
void knn_kernel(const float* __restrict__ q, const float* __restrict__ src,
                int m, int n, int* __restrict__ idx_out, float* __restrict__ d2_out)
{
  __shared__ float sx[256], sy[256], sz[256];
  const int b  = (int)blockIdx.y;
  const int qi = (int)(blockIdx.x * 256 + threadIdx.x);
  const bool valid = qi < m;
  const int ql = valid ? qi : 0;
  const float* qb = q + (long)b * m * 3;
  const float* sb = src + (long)b * n * 3;
  const float qx = qb[ql * 3 + 0], qy = qb[ql * 3 + 1], qz = qb[ql * 3 + 2];

  float bd[KK];
  int   bi[KK];
#pragma unroll
  for (int t = 0; t < KK; t++) { bd[t] = 1e30f; bi[t] = 0; }

  for (int base = 0; base < n; base += 256) {
    const int cnt = (n - base < 256) ? (n - base) : 256;
    __syncthreads();
    if ((int)threadIdx.x < cnt) {
      const float* sp = sb + (long)(base + threadIdx.x) * 3;
      sx[threadIdx.x] = sp[0];
      sy[threadIdx.x] = sp[1];
      sz[threadIdx.x] = sp[2];
    }
    __syncthreads();
    for (int j = 0; j < cnt; j++) {
      const float dx = qx - sx[j], dy = qy - sy[j], dz = qz - sz[j];
      const float d = dx * dx + dy * dy + dz * dz;
      if (d < bd[KK - 1]) {
        bd[KK - 1] = d;
        bi[KK - 1] = base + j;
#pragma unroll
        for (int t = KK - 1; t > 0; t--) {
          if (bd[t] < bd[t - 1]) {
            float td = bd[t]; bd[t] = bd[t - 1]; bd[t - 1] = td;
            int   ti = bi[t]; bi[t] = bi[t - 1]; bi[t - 1] = ti;
          }
        }
      }
    }
  }
  if (valid) {
    int* op = idx_out + ((long)b * m + qi) * KK;
#pragma unroll
    for (int t = 0; t < KK; t++) op[t] = bi[t];
    if (d2_out) {
      float* dp = d2_out + ((long)b * m + qi) * KK;
#pragma unroll
      for (int t = 0; t < KK; t++) dp[t] = bd[t];
    }
  }
}

// ============================================================
// Furthest point sampling (starts at index 0). One block per batch.
// Argmax ties -> lowest index (matches jnp.argmax).
// ============================================================
__global__ __launch_bounds__(256)
void fps_kernel(const float* __restrict__ pts, int n, int m, int* __restrict__ out)
{
  __shared__ float dist[2048];
  __shared__ float rmax[256];
  __shared__ int   rarg[256];
  __shared__ int   s_last;
  const int b = (int)blockIdx.x;
  const float* p = pts + (long)b * n * 3;
  int* o = out + (long)b * m;
  const int tid = (int)threadIdx.x;
  for (int i = tid; i < n; i += 256) dist[i] = 1e10f;
  if (tid == 0) { o[0] = 0; s_last = 0; }
  __syncthreads();
  for (int it = 1; it < m; it++) {
    const int last = s_last;
    const float lx = p[last * 3 + 0], ly = p[last * 3 + 1], lz = p[last * 3 + 2];
    float bestd = -1.f;
    int   besti = 0x7fffffff;
    for (int i = tid; i < n; i += 256) {
      const float dx = p[i * 3 + 0] - lx;
      const float dy = p[i * 3 + 1] - ly;
      const float dz = p[i * 3 + 2] - lz;
      float d = dx * dx + dy * dy + dz * dz;
      const float old = dist[i];
      d = (d < old) ? d : old;
      dist[i] = d;
      if (d > bestd || (d == bestd && i < besti)) { bestd = d; besti = i; }
    }
    rmax[tid] = bestd; rarg[tid] = besti;
    __syncthreads();
    for (int s = 128; s > 0; s >>= 1) {
      if (tid < s) {
        const float od = rmax[tid + s]; const int oi = rarg[tid + s];
        if (od > rmax[tid] || (od == rmax[tid] && oi < rarg[tid])) {
          rmax[tid] = od; rarg[tid] = oi;
        }
      }
      __syncthreads();
    }
    if (tid == 0) { o[it] = rarg[0]; s_last = rarg[0]; }
    __syncthreads();
  }
}

// ---------- gather sampled coordinates: pts [B,n,3] -> out [B,m,3] ----------
__global__ void gather_pts_kernel(const float* __restrict__ pts, const int* __restrict__ fidx,
                                  float* __restrict__ out, int n, int m, int Bn)
{
  const long t = (long)blockIdx.x * blockDim.x + threadIdx.x;
  if (t >= (long)Bn * m) return;
  const int j = (int)(t % m);
  const int b = (int)(t / m);
  const int s = fidx[(long)b * m + j];
  const float* sp = pts + ((long)b * n + s) * 3;
  float* op = out + ((long)b * m + j) * 3;
  op[0] = sp[0]; op[1] = sp[1]; op[2] = sp[2];
}

// ---------- ptsn[b,c,j,k] = pts[b,j,c] - pts[b,idx[b,j,k],c] ----------
__global__ void build_ptsn_kernel(const float* __restrict__ pts, const int* __restrict__ idx,
                                  float* __restrict__ out, int n, int Kk, int Bn)
{
  const long t = (long)blockIdx.x * blockDim.x + threadIdx.x;
  if (t >= (long)Bn * 3 * n * Kk) return;
  const int k = (int)(t % Kk);
  long r = t / Kk;
  const int j = (int)(r % n); r /= n;
  const int c = (int)(r % 3);
  const int b = (int)(r / 3);
  const int src = idx[((long)b * n + j) * Kk + k];
  out[(((long)b * 3 + c) * n + j) * Kk + k] =
      pts[((long)b * n + j) * 3 + c] - pts[((long)b * n + src) * 3 + c];
}

// ---------- xfs[b, m*K+k, j] = x1[b,m,j] - x2[b,m,j,k] + ptf[b,m,j,k] ----------
__global__ void build_xfs_kernel(const float* __restrict__ x1f, const float* __restrict__ x2f,
                                 const float* __restrict__ ptf, float* __restrict__ xfs,
                                 int mid, int Kk, int n, int Bn)
{
  const long t = (long)blockIdx.x * blockDim.x + threadIdx.x;
  if (t >= (long)Bn * mid * n * Kk) return;
  const int k = (int)(t % Kk);
  long r = t / Kk;
  const int j = (int)(r % n); r /= n;
  const int m = (int)(r % mid);
  const int b = (int)(r / mid);
  const long nk = (long)n * Kk;
  const float v = x1f[((long)b * mid + m) * n + j]
                - x2f[((long)b * mid + m) * nk + (long)j * Kk + k]
                + ptf[((long)b * mid + m) * nk + (long)j * Kk + k];
  xfs[(long)b * mid * Kk * n + (long)(m * Kk + k) * n + j] = v;
}

// ---------- softmax over K: w [B, MS*K, n], channel c = ms*K + k ----------
__global__ void softmax_k_kernel(float* __restrict__ w, int MS, int Kk, int n, int Bn)
{
  const long t = (long)blockIdx.x * blockDim.x + threadIdx.x;
  if (t >= (long)Bn * MS * n) return;
  const int j  = (int)(t % n);
  const int ms = (int)((t / n) % MS);
  const int b  = (int)(t / ((long)MS * n));
  float* p = w + ((long)b * MS * Kk + (long)ms * Kk) * n + j;
  float mx = -1e30f;
  for (int k = 0; k < Kk; k++) mx = fmaxf(mx, p[(long)k * n]);
  float e[10];
  float s = 0.f;
  for (int k = 0; k < Kk; k++) { e[k] = expf(p[(long)k * n] - mx); s += e[k]; }
  const float inv = 1.f / s;
  for (int k = 0; k < Kk; k++) p[(long)k * n] = e[k] * inv;
}

// ---------- out[b,m,j] = relu(sum_k w[b, (m%MS)*K+k, j] * x3[b,m,j,k]) ----------
__global__ void attn_out_kernel(const float* __restrict__ wsm, const float* __restrict__ x3s,
                                float* __restrict__ out, int mid, int MS, int Kk, int n, int Bn)
{
  const long t = (long)blockIdx.x * blockDim.x + threadIdx.x;
  if (t >= (long)Bn * mid * n) return;
  const int j = (int)(t % n);
  const int m = (int)((t / n) % mid);
  const int b = (int)(t / ((long)mid * n));
  const int ms = m % MS;
  const float* wp = wsm + ((long)b * MS * Kk + (long)ms * Kk) * n + j;
  const float* xp = x3s + ((long)b * mid + m) * (long)n * Kk + (long)j * Kk;
  float acc = 0.f;
  for (int k = 0; k < Kk; k++) acc += wp[(long)k * n] * xp[k];
  out[((long)b * mid + m) * n + j] = fmaxf(acc, 0.f);
}

// ---------- edge pool: center + max-of-neighbors, channel concat ----------
__global__ void edge_pool_kernel(const float* __restrict__ feat, const int* __restrict__ pidx,
                                 const int* __restrict__ pnidx, float* __restrict__ out,
                                 int C, int n, int m, int Kk, int Bn)
{
  const long t = (long)blockIdx.x * blockDim.x + threadIdx.x;
  if (t >= (long)Bn * C * m) return;
  const int j = (int)(t % m);
  const int c = (int)((t / m) % C);
  const int b = (int)(t / ((long)m * C));
  const float* fb = feat + ((long)b * C + c) * n;
  const float center = fb[pidx[(long)b * m + j]];
  const int* pn = pnidx + ((long)b * m + j) * Kk;
  float mx = -1e30f;
  for (int k = 0; k < Kk; k++) mx = fmaxf(mx, fb[pn[k]]);
  float* ob = out + (long)b * (2 * C) * m;
  ob[(long)c * m + j]       = center;
  ob[(long)(C + c) * m + j] = mx;
}

// ---------- global max over columns: g[o*Bn + b] = max_n x[b,o,n] ----------
__global__ void gmax_kernel(const float* __restrict__ x, float* __restrict__ g,
                            int O, int n, int Bn)
{
  const int t = (int)(blockIdx.x * blockDim.x + threadIdx.x);
  if (t >= Bn * O) return;
  const int b = t % Bn;
  const int o = t / Bn;
  const float* xb = x + ((long)b * O + o) * n;
  float m = -1e30f;
  for (int i = 0; i < n; i++) m = fmaxf(m, xb[i]);
  g[(long)o * Bn + b] = m;
}

// ---------- 3-NN inverse-distance interpolation ----------
__global__ void unpool_interp_kernel(const float* __restrict__ feat, const int* __restrict__ idx,
                                     const float* __restrict__ d2, float* __restrict__ out,
                                     int C, int nsrc, int ntgt, int Bn)
{
  const long t = (long)blockIdx.x * blockDim.x + threadIdx.x;
  if (t >= (long)Bn * C * ntgt) return;
  const int j = (int)(t % ntgt);
  const int c = (int)((t / ntgt) % C);
  const int b = (int)(t / ((long)C * ntgt));
  const int* id = idx + ((long)b * ntgt + j) * 3;
  const float* dd = d2 + ((long)b * ntgt + j) * 3;
  const float w0 = 1.f / fmaxf(dd[0], 1e-10f);
  const float w1 = 1.f / fmaxf(dd[1], 1e-10f);
  const float w2 = 1.f / fmaxf(dd[2], 1e-10f);
  const float inv = 1.f / (w0 + w1 + w2);
  const float* fb = feat + ((long)b * C + c) * nsrc;
  out[((long)b * C + c) * ntgt + j] =
      (fb[id[0]] * w0 + fb[id[1]] * w1 + fb[id[2]] * w2) * inv;
}

// ============================================================
// Host-side orchestration
// ============================================================
static inline int cdiv256(long x) { return (int)((x + 255) / 256); }

static void launch_gemm(hipStream_t s, const float* W, const float* bias, int Ctot,
                        const float* X1, int C1, long sb1, long sc1, long sn1,
                        const float* X2, int C2, long sb2, long sc2, long sn2,
                        const int* gidx, long gsb,
                        float* Y, long ysb, const float* resid, long rsb,
                        int O, int Nc, int Bn, int relu)
{
  const int tiles_n = (Nc + 15) / 16;
  const int tiles_o = (O + 15) / 16;
  const int ngroups = (tiles_n + 7) / 8;
  const int blocks = Bn * tiles_o * ngroups;
  const int useLds = (Ctot <= 1024) ? 1 : 0;          // 16*Ctot*4 <= 64KB
  const size_t shmem = useLds ? (size_t)16 * Ctot * sizeof(float) : 0;
  gemm_wmma_f32<<<dim3(blocks), dim3(256), shmem, s>>>(
      W, bias, Ctot, X1, C1, sb1, sc1, sn1, X2, C2, sb2, sc2, sn2,
      gidx, gsb, Y, ysb, resid, rsb, O, Nc, Bn, relu, useLds);
}

// One PointTransformer block (params at P[pb..pb+14]).
static void run_pt_block(hipStream_t stream, const float* const* P, int pb,
                         int C, int mid, int Np, int Bn,
                         const float* xin, const float* ptsL, const int* idxL,
                         float* tx, float* x1f, float* ptsnB, float* ptf1,
                         float* ptf, float* x2f, float* xfsB, float* hbuf,
                         float* wsmB, float* outm, float* xout)
{
  const int Kn = 10, SH = 8;
  const int MS = mid / SH;
  const long NK = (long)Np * Kn;

  // x = relu(c0 @ xin + b)   (also identity and xa, since relu is idempotent)
  launch_gemm(stream, P[pb + 0], P[pb + 1], C, xin, C, (long)C * Np, Np, 1,
              nullptr, 0, 0, 0, 0, nullptr, 0, tx, (long)C * Np, nullptr, 0, C, Np, Bn, 1);
  // x1 = c1 @ xa + b
  launch_gemm(stream, P[pb + 2], P[pb + 3], C, tx, C, (long)C * Np, Np, 1,
              nullptr, 0, 0, 0, 0, nullptr, 0, x1f, (long)mid * Np, nullptr, 0, mid, Np, Bn, 0);
  // ptsn
  build_ptsn_kernel<<<dim3(cdiv256((long)Bn * 3 * Np * Kn)), dim3(256), 0, stream>>>(
      ptsL, idxL, ptsnB, Np, Kn, Bn);
  // ptf1 = relu(p1 @ ptsn)   (no bias)
  launch_gemm(stream, P[pb + 11], nullptr, 3, ptsnB, 3, 3 * NK, NK, 1,
              nullptr, 0, 0, 0, 0, nullptr, 0, ptf1, (long)mid * NK, nullptr, 0, mid, (int)NK, Bn, 1);
  // ptf = p2 @ ptf1          (no bias)
  launch_gemm(stream, P[pb + 12], nullptr, mid, ptf1, mid, (long)mid * NK, NK, 1,
              nullptr, 0, 0, 0, 0, nullptr, 0, ptf, (long)mid * NK, nullptr, 0, mid, (int)NK, Bn, 0);
  // x2 = c2 @ gather(xa, idx) + b     (fused gather-GEMM, layout [mid][N][K])
  launch_gemm(stream, P[pb + 4], P[pb + 5], C, tx, C, (long)C * Np, Np, 1,
              nullptr, 0, 0, 0, 0, idxL, NK, x2f, (long)mid * NK, nullptr, 0, mid, (int)NK, Bn, 0);
  // x3s = c3 @ gather(xa, idx) + b + ptf   (write into ptf1, now free)
  launch_gemm(stream, P[pb + 6], P[pb + 7], C, tx, C, (long)C * Np, Np, 1,
              nullptr, 0, 0, 0, 0, idxL, NK, ptf1, (long)mid * NK, ptf, (long)mid * NK,
              mid, (int)NK, Bn, 0);
  float* x3s = ptf1;
  // xfs = x1 - x2 + ptf  ->  [B, mid*K, N]
  build_xfs_kernel<<<dim3(cdiv256((long)Bn * mid * Np * Kn)), dim3(256), 0, stream>>>(
      x1f, x2f, ptf, xfsB, mid, Kn, Np, Bn);
  // h = relu(wa @ xfs)
  launch_gemm(stream, P[pb + 8], nullptr, mid * Kn, xfsB, mid * Kn, (long)mid * Kn * Np, Np, 1,
              nullptr, 0, 0, 0, 0, nullptr, 0, hbuf, (long)MS * Np, nullptr, 0, MS, Np, Bn, 1);
  // w = wb @ h + b
  launch_gemm(stream, P[pb + 9], P[pb + 10], MS, hbuf, MS, (long)MS * Np, Np, 1,
              nullptr, 0, 0, 0, 0, nullptr, 0, wsmB, (long)MS * Kn * Np, nullptr, 0,
              MS * Kn, Np, Bn, 0);
  // softmax over K
  softmax_k_kernel<<<dim3(cdiv256((long)Bn * MS * Np)), dim3(256), 0, stream>>>(
      wsmB, MS, Kn, Np, Bn);
  // out = relu(sum_k w * x3s)
  attn_out_kernel<<<dim3(cdiv256((long)Bn * mid * Np)), dim3(256), 0, stream>>>(
      wsmB, x3s, outm, mid, MS, Kn, Np, Bn);
  // xout = relu(co @ out + b + identity)    (outer relu fused)
  launch_gemm(stream, P[pb + 13], P[pb + 14], mid, outm, mid, (long)mid * Np, Np, 1,
              nullptr, 0, 0, 0, 0, nullptr, 0, xout, (long)C * Np, tx, (long)C * Np, C, Np, Bn, 1);
}

extern "C" void kernel_launch(void* const* d_in, const int* in_sizes, int n_in,
                              void* d_out, int out_size, void* d_ws, size_t ws_size,
                              hipStream_t stream)
{
  (void)in_sizes; (void)out_size; (void)ws_size;
  const int B = 8, N1 = 2048, N2 = 512, N3 = 128, Kn = 10;

  const float* features = (const float*)d_in[0];   // [B,3,N1]
  const float* pts      = (const float*)d_in[1];   // [B,N1,3]
  const float* P[64];
  for (int i = 2; i < n_in && i < 64; i++) P[i] = (const float*)d_in[i];
  // param indices: conv0=2,3 | PT1=4..18 | PT2=19..33 | PT3=34..48
  // conv_g=49,50 | fc1=51,52 | fc2=53,54 | conv3=55,56 | conv2=57,58
  // conv1=59,60 | conv_out=61,62

  size_t off = 0;
  auto alloc = [&](size_t nfloats) -> float* {
    float* p = (float*)d_ws + off;
    off += (nfloats + 63) & ~(size_t)63;
    return p;
  };

  float* x0    = alloc((size_t)B * 64 * N1);
  float* x1    = alloc((size_t)B * 64 * N1);
  float* pool1 = alloc((size_t)B * 128 * N2);
  float* x2    = alloc((size_t)B * 128 * N2);
  float* pool2 = alloc((size_t)B * 256 * N3);
  float* x3    = alloc((size_t)B * 256 * N3);
  float* pt2   = alloc((size_t)B * N2 * 3);
  float* pt3   = alloc((size_t)B * N3 * 3);
  int* idx1  = (int*)alloc((size_t)B * N1 * Kn);
  int* idx2  = (int*)alloc((size_t)B * N2 * Kn);
  int* idx3  = (int*)alloc((size_t)B * N3 * Kn);
  int* fidx1 = (int*)alloc((size_t)B * N2);
  int* fidx2 = (int*)alloc((size_t)B * N3);
  int* pn1   = (int*)alloc((size_t)B * N2 * Kn);
  int* pn2   = (int*)alloc((size_t)B * N3 * Kn);
  // PT scratch, sized for PT1 (largest): mid*N and mid*N*K shrink at deeper levels
  float* tx    = alloc((size_t)B * 256 * N1);          // C*N  (generous)
  float* x1f   = alloc((size_t)B * 16 * N1);           // mid*N
  float* ptsnB = alloc((size_t)B * 3 * N1 * Kn);       // 3*N*K
  float* ptf1  = alloc((size_t)B * 16 * N1 * Kn);      // mid*N*K
  float* ptf   = alloc((size_t)B * 16 * N1 * Kn);
  float* x2f   = alloc((size_t)B * 16 * N1 * Kn);
  float* xfsB  = alloc((size_t)B * 16 * Kn * N1);
  float* hbuf  = alloc((size_t)B * 8 * N1);            // (mid/8)*N
  float* wsmB  = alloc((size_t)B * 2 * Kn * N1);       // (mid/8)*K*N
  float* outm  = alloc((size_t)B * 16 * N1);           // mid*N
  float* convg = alloc((size_t)B * 1024 * N3);
  float* g0    = alloc((size_t)1024 * B);
  float* g1    = alloc((size_t)512 * B);
  float* g2    = alloc((size_t)1024 * B);
  float* c3out = alloc((size_t)B * 256 * N3);
  int*   upi2  = (int*)alloc((size_t)B * N2 * 3);
  float* upd2b = alloc((size_t)B * N2 * 3);
  float* up2   = alloc((size_t)B * 256 * N2);
  float* c2out = alloc((size_t)B * 128 * N2);
  int*   upi1  = (int*)alloc((size_t)B * N1 * 3);
  float* upd1b = alloc((size_t)B * N1 * 3);
  float* up1   = alloc((size_t)B * 128 * N1);
  float* c1out = alloc((size_t)B * 128 * N1);

  // ---- conv0: x0 = W0 @ features + b (no relu) ----
  launch_gemm(stream, P[2], P[3], 3, features, 3, (long)3 * N1, N1, 1,
              nullptr, 0, 0, 0, 0, nullptr, 0, x0, (long)64 * N1, nullptr, 0, 64, N1, B, 0);

  // ---- level 1 ----
  knn_kernel<10><<<dim3((N1 + 255) / 256, B), dim3(256), 0, stream>>>(pts, pts, N1, N1, idx1, nullptr);
  run_pt_block(stream, P, 4, 64, 16, N1, B, x0, pts, idx1,
               tx, x1f, ptsnB, ptf1, ptf, x2f, xfsB, hbuf, wsmB, outm, x1);

  // ---- edge pool 1: N1 -> N2, C 64 -> 128 ----
  fps_kernel<<<dim3(B), dim3(256), 0, stream>>>(pts, N1, N2, fidx1);
  gather_pts_kernel<<<dim3(cdiv256((long)B * N2)), dim3(256), 0, stream>>>(pts, fidx1, pt2, N1, N2, B);
  knn_kernel<10><<<dim3((N2 + 255) / 256, B), dim3(256), 0, stream>>>(pt2, pts, N2, N1, pn1, nullptr);
  edge_pool_kernel<<<dim3(cdiv256((long)B * 64 * N2)), dim3(256), 0, stream>>>(
      x1, fidx1, pn1, pool1, 64, N1, N2, Kn, B);

  // ---- level 2 ----
  knn_kernel<10><<<dim3((N2 + 255) / 256, B), dim3(256), 0, stream>>>(pt2, pt2, N2, N2, idx2, nullptr);
  run_pt_block(stream, P, 19, 128, 32, N2, B, pool1, pt2, idx2,
               tx, x1f, ptsnB, ptf1, ptf, x2f, xfsB, hbuf, wsmB, outm, x2);

  // ---- edge pool 2: N2 -> N3, C 128 -> 256 ----
  fps_kernel<<<dim3(B), dim3(256), 0, stream>>>(pt2, N2, N3, fidx2);
  gather_pts_kernel<<<dim3(cdiv256((long)B * N3)), dim3(256), 0, stream>>>(pt2, fidx2, pt3, N2, N3, B);
  knn_kernel<10><<<dim3((N3 + 255) / 256, B), dim3(256), 0, stream>>>(pt3, pt2, N3, N2, pn2, nullptr);
  edge_pool_kernel<<<dim3(cdiv256((long)B * 128 * N3)), dim3(256), 0, stream>>>(
      x2, fidx2, pn2, pool2, 128, N2, N3, Kn, B);

  // ---- level 3 ----
  knn_kernel<10><<<dim3((N3 + 255) / 256, B), dim3(256), 0, stream>>>(pt3, pt3, N3, N3, idx3, nullptr);
  run_pt_block(stream, P, 34, 256, 64, N3, B, pool2, pt3, idx3,
               tx, x1f, ptsnB, ptf1, ptf, x2f, xfsB, hbuf, wsmB, outm, x3);

  // ---- global branch ----
  launch_gemm(stream, P[49], P[50], 256, x3, 256, (long)256 * N3, N3, 1,
              nullptr, 0, 0, 0, 0, nullptr, 0, convg, (long)1024 * N3, nullptr, 0, 1024, N3, B, 0);
  gmax_kernel<<<dim3(cdiv256((long)B * 1024)), dim3(256), 0, stream>>>(convg, g0, 1024, N3, B);
  // fc1: batch-as-columns (layout [O][B]); relu
  launch_gemm(stream, P[51], P[52], 1024, g0, 1024, 0, B, 1,
              nullptr, 0, 0, 0, 0, nullptr, 0, g1, (long)512 * B, nullptr, 0, 512, B, 1, 1);
  // fc2; relu
  launch_gemm(stream, P[53], P[54], 512, g1, 512, 0, B, 1,
              nullptr, 0, 0, 0, 0, nullptr, 0, g2, (long)1024 * B, nullptr, 0, 1024, B, 1, 1);

  // ---- conv3: relu(W @ concat(gx_broadcast, x3)) ----
  launch_gemm(stream, P[55], P[56], 1280,
              g2, 1024, 1, B, 0,                 // broadcast over columns (sn=0), batch stride 1
              x3, 256, (long)256 * N3, N3, 1,
              nullptr, 0, c3out, (long)256 * N3, nullptr, 0, 256, N3, B, 1);

  // ---- unpool N3 -> N2 ----
  knn_kernel<3><<<dim3((N2 + 255) / 256, B), dim3(256), 0, stream>>>(pt2, pt3, N2, N3, upi2, upd2b);
  unpool_interp_kernel<<<dim3(cdiv256((long)B * 256 * N2)), dim3(256), 0, stream>>>(
      c3out, upi2, upd2b, up2, 256, N3, N2, B);

  // ---- conv2: relu(W @ concat(up2, x2)) ----
  launch_gemm(stream, P[57], P[58], 384,
              up2, 256, (long)256 * N2, N2, 1,
              x2, 128, (long)128 * N2, N2, 1,
              nullptr, 0, c2out, (long)128 * N2, nullptr, 0, 128, N2, B, 1);

  // ---- unpool N2 -> N1 ----
  knn_kernel<3><<<dim3((N1 + 255) / 256, B), dim3(256), 0, stream>>>(pts, pt2, N1, N2, upi1, upd1b);
  unpool_interp_kernel<<<dim3(cdiv256((long)B * 128 * N1)), dim3(256), 0, stream>>>(
      c2out, upi1, upd1b, up1, 128, N2, N1, B);

  // ---- conv1: relu(W @ concat(up1, x1)) ----
  launch_gemm(stream, P[59], P[60], 192,
              up1, 128, (long)128 * N1, N1, 1,
              x1, 64, (long)64 * N1, N1, 1,
              nullptr, 0, c1out, (long)128 * N1, nullptr, 0, 128, N1, B, 1);

  // ---- conv_out: W @ concat(c1out, x0) -> d_out [B,128,N1] ----
  launch_gemm(stream, P[61], P[62], 192,
              c1out, 128, (long)128 * N1, N1, 1,
              x0, 64, (long)64 * N1, N1, 1,
              nullptr, 0, (float*)d_out, (long)128 * N1, nullptr, 0, 128, N1, B, 0);
}